// IDM_SGC_23390391894805
// MI455X (gfx1250) — compile-verified
//
#include <hip/hip_runtime.h>
#include <math.h>

// ---------------------------------------------------------------------------
// EIGNN closed-form solve via Neumann series in Q_S's eigenbasis, with
// logarithmic "doubling" of the partial sum:
//   Z = [ sum_{k>=0} (g*gF)^k (X Q_S) diag(L)^k ] Q_S^T
//   Z_{p+1} = Z_p + A_p Z_p diag(L_p);  A_{p+1}=A_p^2; L_{p+1}=L_p^2
// 7 doubling steps = 128 series terms (error ~0.792^128 ~ 1e-13).
// All heavy math is f32 WMMA (v_wmma_f32_16x16x4_f32) to match f32 reference.
// ---------------------------------------------------------------------------

typedef float v2f __attribute__((ext_vector_type(2)));
typedef float v8f __attribute__((ext_vector_type(8)));

#define GAMMA_C 0.8f
#define EPS_C   1e-12f

constexpr int BM = 64;         // block tile M
constexpr int BN = 128;        // block tile N
constexpr int KB = 32;         // K chunk staged in LDS (8 wmma k-steps)
constexpr int ASTR = 36;       // LDS row stride for As[m][k] (16B rows, no bank conflicts)
constexpr int BP = 144;        // float2-row stride for B k-pair layout (halves hit disjoint banks)
constexpr int STEPS = 7;       // doublings -> 2^7 = 128 terms

// MODE 0: Cout  = A*B
// MODE 1: Cout = Cin + lam[col] * (A*B)    (doubling step; Cin read-only)
template <int MODE, bool AT, bool BT>
__global__ __launch_bounds__(256) void gemm_wmma_f32(
    const float* __restrict__ Asrc, const float* __restrict__ Bsrc,
    const float* __restrict__ Cin, float* __restrict__ Cout,
    const float* __restrict__ lam,
    int M, int N, int K, int lda, int ldb) {
  __shared__ __align__(16) float As[BM * ASTR];          // As[m][k]
  __shared__ __align__(16) float Bs[(KB / 2) * BP * 2];  // float2 pairs: (kpair, n)

  const int t    = threadIdx.x;
  const int wave = t >> 5;
  const int lane = t & 31;
  const int half = lane >> 4;   // lanes 16-31: K+2 (A/B frags), M+8 (C/D)
  const int lr   = lane & 15;
  const int wm   = wave & 3;    // 4 wave-rows of 16
  const int wn   = wave >> 2;   // 2 wave-cols of 64

  const int tileM = blockIdx.y * BM;
  const int tileN = blockIdx.x * BN;

  v8f c[4];
#pragma unroll
  for (int i = 0; i < 4; ++i) c[i] = (v8f){0.f, 0.f, 0.f, 0.f, 0.f, 0.f, 0.f, 0.f};

  for (int kb = 0; kb < K; kb += KB) {
    // ---- stage A tile As[m][k] (64x32) ----
    if constexpr (!AT) {
#pragma unroll
      for (int r = 0; r < 2; ++r) {
        const int idx = t + r * 256;
        const int m  = idx >> 3;
        const int kq = (idx & 7) << 2;
        const float4 v = *reinterpret_cast<const float4*>(
            &Asrc[(size_t)(tileM + m) * lda + kb + kq]);
        *reinterpret_cast<float4*>(&As[m * ASTR + kq]) = v;
      }
    } else {  // A[m][k] = Asrc[k][m]  (F^T F)
#pragma unroll
      for (int r = 0; r < 2; ++r) {
        const int idx = t + r * 256;
        const int k  = idx >> 4;
        const int mq = (idx & 15) << 2;
        const float4 v = *reinterpret_cast<const float4*>(
            &Asrc[(size_t)(kb + k) * lda + tileM + mq]);
        As[(mq + 0) * ASTR + k] = v.x;
        As[(mq + 1) * ASTR + k] = v.y;
        As[(mq + 2) * ASTR + k] = v.z;
        As[(mq + 3) * ASTR + k] = v.w;
      }
    }
    // ---- stage B tile as k-pairs: Bs[(k>>1)*BP + n] = {B[k][n], B[k+1][n]} ----
    if constexpr (!BT) {
#pragma unroll
      for (int r = 0; r < 4; ++r) {
        const int idx = t + r * 256;
        const int k  = idx >> 5;          // 0..31
        const int nq = (idx & 31) << 2;   // 0..124
        const float4 v = *reinterpret_cast<const float4*>(
            &Bsrc[(size_t)(kb + k) * ldb + tileN + nq]);
        const int base = ((k >> 1) * BP + nq) * 2 + (k & 1);
        Bs[base + 0] = v.x;
        Bs[base + 2] = v.y;
        Bs[base + 4] = v.z;
        Bs[base + 6] = v.w;
      }
    } else {  // B[k][n] = Bsrc[n][k]  (acc @ Q_S^T): contiguous k -> float2 stores
#pragma unroll
      for (int r = 0; r < 4; ++r) {
        const int idx = t + r * 256;
        const int n  = idx >> 3;          // 0..127
        const int kq = idx & 7;           // quad of 4 k's -> 2 pairs
        const float4 v = *reinterpret_cast<const float4*>(
            &Bsrc[(size_t)(tileN + n) * ldb + kb + kq * 4]);
        const int p0 = kq * 2;
        *reinterpret_cast<v2f*>(&Bs[((p0 + 0) * BP + n) * 2]) = (v2f){v.x, v.y};
        *reinterpret_cast<v2f*>(&Bs[((p0 + 1) * BP + n) * 2]) = (v2f){v.z, v.w};
      }
    }
    __syncthreads();

#pragma unroll
    for (int kc = 0; kc < KB / 4; ++kc) {
      const int kk    = kc * 4 + half * 2;  // A/B fragment K base for this half
      const int kpair = kc * 2 + half;
      const v2f a = *reinterpret_cast<const v2f*>(&As[(wm * 16 + lr) * ASTR + kk]);
#pragma unroll
      for (int nt = 0; nt < 4; ++nt) {
        const int n = wn * 64 + nt * 16 + lr;
        const v2f b = *reinterpret_cast<const v2f*>(&Bs[(kpair * BP + n) * 2]);
        c[nt] = __builtin_amdgcn_wmma_f32_16x16x4_f32(
            false, a, false, b, (short)0, c[nt], false, false);
      }
    }
    __syncthreads();
  }

  // epilogue: C layout -> VGPR r holds M = r + 8*half, N = lr
#pragma unroll
  for (int nt = 0; nt < 4; ++nt) {
    const int col = tileN + wn * 64 + nt * 16 + lr;
    float lv = 1.0f;
    if constexpr (MODE == 1) lv = lam[col];
#pragma unroll
    for (int r = 0; r < 8; ++r) {
      const int row = tileM + wm * 16 + half * 8 + r;
      const size_t idx = (size_t)row * N + col;
      if constexpr (MODE == 0) {
        Cout[idx] = c[nt][r];
      } else {
        Cout[idx] = Cin[idx] + lv * c[nt][r];
      }
    }
  }
}

__global__ void zero_kernel(float* p) {
  if (threadIdx.x == 0 && blockIdx.x == 0) p[0] = 0.0f;
}

__global__ __launch_bounds__(256) void frob_norm_sq(
    const float* __restrict__ x, int n, float* __restrict__ out) {
  __shared__ float s[256];
  float acc = 0.0f;
  for (int i = blockIdx.x * blockDim.x + threadIdx.x; i < n;
       i += gridDim.x * blockDim.x) {
    const float v = x[i];
    acc += v * v;
  }
  s[threadIdx.x] = acc;
  __syncthreads();
  for (int off = 128; off > 0; off >>= 1) {
    if (threadIdx.x < (unsigned)off) s[threadIdx.x] += s[threadIdx.x + off];
    __syncthreads();
  }
  if (threadIdx.x == 0) atomicAdd(out, s[0]);
}

// y = (gamma / (||FtF||_F + eps)) * x
__global__ __launch_bounds__(256) void scale_kernel(
    const float* __restrict__ x, float* __restrict__ y,
    const float* __restrict__ nrm_sq, int n) {
  const float s = GAMMA_C / (sqrtf(nrm_sq[0]) + EPS_C);
  for (int i = blockIdx.x * blockDim.x + threadIdx.x; i < n;
       i += gridDim.x * blockDim.x)
    y[i] = x[i] * s;
}

__global__ __launch_bounds__(256) void square_kernel(
    const float* __restrict__ x, float* __restrict__ y, int n) {
  for (int i = blockIdx.x * blockDim.x + threadIdx.x; i < n;
       i += gridDim.x * blockDim.x)
    y[i] = x[i] * x[i];
}

extern "C" void kernel_launch(void* const* d_in, const int* in_sizes, int n_in,
                              void* d_out, int out_size, void* d_ws,
                              size_t ws_size, hipStream_t stream) {
  const float* X   = (const float*)d_in[0];  // (m, n)
  const float* F   = (const float*)d_in[1];  // (m, m)
  const float* Q   = (const float*)d_in[2];  // (n, n)
  const float* Lam = (const float*)d_in[3];  // (n,)
  float* out = (float*)d_out;                // (m, n)

  const int n = in_sizes[3];
  const int m = in_sizes[0] / n;

  float* ws  = (float*)d_ws;
  float* FtF = ws;                           // m*m
  float* Ap0 = FtF + (size_t)m * m;          // m*m  (A_0 = gamma*gF)
  float* Ap1 = Ap0 + (size_t)m * m;          // m*m
  float* Z0  = Ap1 + (size_t)m * m;          // m*n
  float* Z1  = Z0 + (size_t)m * n;           // m*n
  float* L0  = Z1 + (size_t)m * n;           // n
  float* L1  = L0 + (size_t)n;               // n
  float* nrm = L1 + (size_t)n;               // 1

  const dim3 gridBig(n / BN, m / BM);        // (32, 8)
  const dim3 gridSml(m / BN, m / BM);        // (4, 8)

  // gF = F^T F / (||F^T F||_F + eps), folded with gamma into A_0
  zero_kernel<<<1, 32, 0, stream>>>(nrm);
  gemm_wmma_f32<0, true, false><<<gridSml, 256, 0, stream>>>(
      F, F, nullptr, FtF, nullptr, m, m, m, m, m);
  frob_norm_sq<<<dim3(256), 256, 0, stream>>>(FtF, m * m, nrm);
  scale_kernel<<<dim3(256), 256, 0, stream>>>(FtF, Ap0, nrm, m * m);

  // Z_0 = X @ Q_S
  gemm_wmma_f32<0, false, false><<<gridBig, 256, 0, stream>>>(
      X, Q, nullptr, Z0, nullptr, m, n, n, n, n);

  // doubling: Z_{p+1} = Z_p + A_p Z_p diag(L_p);  A_{p+1}=A_p^2; L_{p+1}=L_p^2
  float* Zold = Z0; float* Znew = Z1;
  float* Acur = Ap0; float* Anew = Ap1;
  const float* Lcur = Lam;
  float* Lbufs[2] = {L0, L1};
  for (int p = 0; p < STEPS; ++p) {
    gemm_wmma_f32<1, false, false><<<gridBig, 256, 0, stream>>>(
        Acur, Zold, Zold, Znew, Lcur, m, n, m, m, n);
    if (p < STEPS - 1) {
      gemm_wmma_f32<0, false, false><<<gridSml, 256, 0, stream>>>(
          Acur, Acur, nullptr, Anew, nullptr, m, m, m, m, m);
      square_kernel<<<dim3(16), 256, 0, stream>>>(Lcur, Lbufs[p & 1], n);
      Lcur = Lbufs[p & 1];
      float* ta = Acur; Acur = Anew; Anew = ta;
    }
    float* tz = Zold; Zold = Znew; Znew = tz;
  }

  // Z = Z_final @ Q_S^T
  gemm_wmma_f32<0, false, true><<<gridBig, 256, 0, stream>>>(
      Zold, Q, nullptr, out, nullptr, m, n, n, n, n);
}